// RDM_Module_48249662603860
// MI455X (gfx1250) — compile-verified
//
#include <hip/hip_runtime.h>

typedef __attribute__((ext_vector_type(2))) float v2f;
typedef __attribute__((ext_vector_type(4))) float v4f;
typedef __attribute__((ext_vector_type(8))) float v8f;

#define NMAT   4096
#define BM     128
#define BN     128
#define BK     32
#define LDST   (BK + 4)          // 36 dwords row stride: 16B-aligned rows, conflict-free frag reads

// ---------------------------------------------------------------------------
// Kernel 1: rowsq[r] = sum_k E[r][k]^2
// ---------------------------------------------------------------------------
__global__ __launch_bounds__(256)
void row_norms_kernel(const float* __restrict__ E, float* __restrict__ sq) {
    __shared__ float red[256];
    const int row = blockIdx.x;
    const float* e = E + (size_t)row * NMAT;
    float s = 0.0f;
    for (int c = threadIdx.x; c < NMAT; c += 256) {
        float v = e[c];
        s += v * v;
    }
    red[threadIdx.x] = s;
    __syncthreads();
    for (int off = 128; off > 0; off >>= 1) {
        if (threadIdx.x < (unsigned)off) red[threadIdx.x] += red[threadIdx.x + off];
        __syncthreads();
    }
    if (threadIdx.x == 0) sq[row] = red[0];
}

// ---------------------------------------------------------------------------
// Kernel 2: D[i][j] = max(sq[i] + sq[j] - 2 * (E E^T)[i][j], 0)
// 256 threads = 8 waves (2x4); each wave computes a 64x32 sub-tile as
// 4x2 WMMA 16x16 accumulators using v_wmma_f32_16x16x4_f32.
// ---------------------------------------------------------------------------
__global__ __launch_bounds__(256)
void dist_wmma_kernel(const float* __restrict__ E,
                      const float* __restrict__ sq,
                      float* __restrict__ out) {
    __shared__ float lA[BM * LDST];
    __shared__ float lB[BN * LDST];

    const int tid   = threadIdx.x;
    const int lane  = tid & 31;
    const int wave  = tid >> 5;
    const int wm    = wave >> 2;          // 0..1  -> 64-row band within block tile
    const int wn    = wave & 3;           // 0..3  -> 32-col band
    const int m16   = lane & 15;          // row/col within 16x16 fragment
    const int khalf = lane >> 4;          // 0/1   -> K pair (0,1) vs (2,3)

    const int iBase = blockIdx.y * BM;
    const int jBase = blockIdx.x * BN;

    const float* gA = E + (size_t)iBase * NMAT;   // 128 rows for output rows
    const float* gB = E + (size_t)jBase * NMAT;   // 128 rows for output cols

    v8f acc[4][2];
    const v8f vzero = {0.f, 0.f, 0.f, 0.f, 0.f, 0.f, 0.f, 0.f};
#pragma unroll
    for (int tm = 0; tm < 4; ++tm)
#pragma unroll
        for (int tn = 0; tn < 2; ++tn) acc[tm][tn] = vzero;

    // Per-thread global->register staging: 4 float4 per slab per K-chunk.
    v4f pa[4], pb[4];

    auto load_regs = [&](int kc) {
#pragma unroll
        for (int i = 0; i < 4; ++i) {
            const int l = tid + i * 256;          // 0..1023
            const int r = l >> 3;                 // row 0..127
            const int c = (l & 7) << 2;           // col 0,4,...,28
            pa[i] = *(const v4f*)(gA + (size_t)r * NMAT + kc + c);
            pb[i] = *(const v4f*)(gB + (size_t)r * NMAT + kc + c);
        }
    };
    auto store_lds = [&]() {
#pragma unroll
        for (int i = 0; i < 4; ++i) {
            const int l = tid + i * 256;
            const int r = l >> 3;
            const int c = (l & 7) << 2;
            *(v4f*)(&lA[r * LDST + c]) = pa[i];
            *(v4f*)(&lB[r * LDST + c]) = pb[i];
        }
    };

    load_regs(0);
    store_lds();
    __syncthreads();

    const int aRow = wm * 64 + m16;
    const int bRow = wn * 32 + m16;

    for (int kc = 0; kc < NMAT; kc += BK) {
        const bool more = (kc + BK) < NMAT;
        if (more) load_regs(kc + BK);             // overlap global latency with WMMA

#pragma unroll
        for (int k = 0; k < BK; k += 4) {
            v2f af[4], bf[2];
#pragma unroll
            for (int tm = 0; tm < 4; ++tm)
                af[tm] = *(const v2f*)(&lA[(aRow + tm * 16) * LDST + k + khalf * 2]);
#pragma unroll
            for (int tn = 0; tn < 2; ++tn)
                bf[tn] = *(const v2f*)(&lB[(bRow + tn * 16) * LDST + k + khalf * 2]);
#pragma unroll
            for (int tm = 0; tm < 4; ++tm)
#pragma unroll
                for (int tn = 0; tn < 2; ++tn)
                    acc[tm][tn] = __builtin_amdgcn_wmma_f32_16x16x4_f32(
                        false, af[tm], false, bf[tn],
                        (short)0, acc[tm][tn], false, false);
        }

        __syncthreads();                          // done reading LDS
        if (more) store_lds();
        __syncthreads();                          // LDS ready for next chunk
    }

    // Fused epilogue: D = max(sq[i] + sq[j] - 2*acc, 0)
    const int iW = iBase + wm * 64;
    const int jW = jBase + wn * 32;
#pragma unroll
    for (int tn = 0; tn < 2; ++tn) {
        const int j    = jW + tn * 16 + m16;
        const float sqj = sq[j];
#pragma unroll
        for (int tm = 0; tm < 4; ++tm) {
            const int i0 = iW + tm * 16 + khalf * 8;  // C/D layout: M = r + 8*(lane/16)
#pragma unroll
            for (int r = 0; r < 8; ++r) {
                const float sqi = sq[i0 + r];
                float v = sqi + sqj - 2.0f * acc[tm][tn][r];
                v = fmaxf(v, 0.0f);
                out[(size_t)(i0 + r) * NMAT + j] = v;
            }
        }
    }
}

// ---------------------------------------------------------------------------
extern "C" void kernel_launch(void* const* d_in, const int* in_sizes, int n_in,
                              void* d_out, int out_size, void* d_ws, size_t ws_size,
                              hipStream_t stream) {
    (void)in_sizes; (void)n_in; (void)out_size; (void)ws_size;
    const float* E  = (const float*)d_in[0];
    float*       out = (float*)d_out;
    float*       sq  = (float*)d_ws;              // 4096 floats = 16 KB scratch

    row_norms_kernel<<<NMAT, 256, 0, stream>>>(E, sq);

    dim3 grid(NMAT / BN, NMAT / BM);              // 32 x 32 workgroups
    dist_wmma_kernel<<<grid, 256, 0, stream>>>(E, sq, out);
}